// Body3_L_14139032339084
// MI455X (gfx1250) — compile-verified
//
#include <hip/hip_runtime.h>
#include <cstdint>
#include <math.h>

#define BS 256   // threads per block == samples per block (12 KB in, 12 KB out per block)

// ---- CDNA5 async global<->LDS path detection -------------------------------
#if defined(__has_builtin)
# if __has_builtin(__builtin_amdgcn_global_load_async_to_lds_b128) && \
     __has_builtin(__builtin_amdgcn_global_store_async_from_lds_b128)
#  define HAVE_ASYNC_LDS 1
# endif
#endif

// Builtin parameter types (from hipcc diagnostic): int4-vector pointers,
// global (AS1) for the memory side, LDS (AS3) for the LDS side.
typedef int v4i __attribute__((vector_size(16)));
typedef __attribute__((address_space(1))) v4i gv4i;
typedef __attribute__((address_space(3))) v4i lv4i;

__device__ __forceinline__ void wait_async_zero() {
#if defined(__has_builtin) && __has_builtin(__builtin_amdgcn_s_wait_asynccnt)
  __builtin_amdgcn_s_wait_asynccnt(0);
#else
  asm volatile("s_wait_asynccnt 0" ::: "memory");
#endif
}

// ---- closed-form Euler-Lagrange for this system ----------------------------
// dvdvL = I, dxdvL = 0  =>  a = dxL:
// a_i = sum_{j != i} (x_j - x_i) / |x_j - x_i|^3   (k = 1, m = 1)
__device__ __forceinline__ void body3_compute(const float c[12], float o[12]) {
  const float p0x = c[0], p0y = c[1];
  const float p1x = c[2], p1y = c[3];
  const float p2x = c[4], p2y = c[5];
  float a0x = 0.f, a0y = 0.f, a1x = 0.f, a1y = 0.f, a2x = 0.f, a2y = 0.f;
  {
    const float dx = p0x - p1x, dy = p0y - p1y;
    const float r2 = dx * dx + dy * dy;
    const float i3 = 1.0f / (r2 * sqrtf(r2));
    const float fx = dx * i3, fy = dy * i3;
    a0x -= fx; a0y -= fy; a1x += fx; a1y += fy;
  }
  {
    const float dx = p0x - p2x, dy = p0y - p2y;
    const float r2 = dx * dx + dy * dy;
    const float i3 = 1.0f / (r2 * sqrtf(r2));
    const float fx = dx * i3, fy = dy * i3;
    a0x -= fx; a0y -= fy; a2x += fx; a2y += fy;
  }
  {
    const float dx = p1x - p2x, dy = p1y - p2y;
    const float r2 = dx * dx + dy * dy;
    const float i3 = 1.0f / (r2 * sqrtf(r2));
    const float fx = dx * i3, fy = dy * i3;
    a1x -= fx; a1y -= fy; a2x += fx; a2y += fy;
  }
  // out = [v (6), a (6)]
  o[0] = c[6];  o[1] = c[7];  o[2] = c[8];
  o[3] = c[9];  o[4] = c[10]; o[5] = c[11];
  o[6] = a0x;   o[7] = a0y;   o[8] = a1x;
  o[9] = a1y;   o[10] = a2x;  o[11] = a2y;
}

__global__ __launch_bounds__(BS) void body3_kernel(const float* __restrict__ coords,
                                                   float* __restrict__ out, int n) {
#if defined(HAVE_ASYNC_LDS)
  __shared__ __align__(16) float sIn [BS * 12];
  __shared__ __align__(16) float sOut[BS * 12];
#endif
  const int tid  = threadIdx.x;
  const int base = blockIdx.x * BS;
  const bool full = (base + BS) <= n;

#if defined(HAVE_ASYNC_LDS)
  if (full) {
    const float* gin  = coords + (size_t)base * 12;
    float*       gout = out    + (size_t)base * 12;

    // Stage 12 KB input tile: 3 rounds of per-lane 16B async copies, wave-linear.
#pragma unroll
    for (int r = 0; r < 3; ++r) {
      const int off = (r * BS + tid) * 16;  // byte offset, contiguous across lanes
      char* gsrc = const_cast<char*>(reinterpret_cast<const char*>(gin)) + off;
      __builtin_amdgcn_global_load_async_to_lds_b128(
          (gv4i*)gsrc, (lv4i*)((char*)sIn + off), 0, 0);
    }
    wait_async_zero();
    __syncthreads();

    float c[12], o[12];
    {
      const float4* s4 = (const float4*)(&sIn[tid * 12]);  // 48B-aligned per lane
      ((float4*)c)[0] = s4[0];
      ((float4*)c)[1] = s4[1];
      ((float4*)c)[2] = s4[2];
    }
    body3_compute(c, o);
    {
      float4* d4 = (float4*)(&sOut[tid * 12]);
      d4[0] = ((float4*)o)[0];
      d4[1] = ((float4*)o)[1];
      d4[2] = ((float4*)o)[2];
    }
    __syncthreads();

    // Drain 12 KB output tile with async LDS->global stores, wave-linear.
#pragma unroll
    for (int r = 0; r < 3; ++r) {
      const int off = (r * BS + tid) * 16;
      __builtin_amdgcn_global_store_async_from_lds_b128(
          (gv4i*)((char*)gout + off), (lv4i*)((char*)sOut + off), 0, 0);
    }
    wait_async_zero();
    return;
  }
#endif

  // Fallback / tail path: direct b128 global traffic.
  const int idx = base + tid;
  if (idx < n) {
    float c[12], o[12];
    const float4* g4 = (const float4*)(coords + (size_t)idx * 12);
    ((float4*)c)[0] = g4[0];
    ((float4*)c)[1] = g4[1];
    ((float4*)c)[2] = g4[2];
    body3_compute(c, o);
    float4* q4 = (float4*)(out + (size_t)idx * 12);
    q4[0] = ((float4*)o)[0];
    q4[1] = ((float4*)o)[1];
    q4[2] = ((float4*)o)[2];
  }
}

extern "C" void kernel_launch(void* const* d_in, const int* in_sizes, int n_in,
                              void* d_out, int out_size, void* d_ws, size_t ws_size,
                              hipStream_t stream) {
  // inputs: d_in[0] = t (1 elem, unused), d_in[1] = coords [bs, 12] f32
  const float* coords = (const float*)d_in[1];
  float*       out    = (float*)d_out;
  const int n = in_sizes[1] / 12;               // 524288 samples
  const int grid = (n + BS - 1) / BS;           // 2048 blocks of 256 (8 waves @ wave32)
  body3_kernel<<<grid, BS, 0, stream>>>(coords, out, n);
}